// DeconvCG_52888227283269
// MI455X (gfx1250) — compile-verified
//
#include <hip/hip_runtime.h>

// DeconvCG on MI455X (gfx1250, wave32).
// 15x15 depthwise correlation cast as banded GEMM on V_WMMA_F32_16X16X4_F32:
//   one wave per 16x16 output tile; A = LDS input patch rows, B = banded shift
//   matrix built from a zero-padded PSF row buffer (no bounds checks -> no
//   exec-masked loads); 15 (dy) x 8 (column chunks of 4) WMMAs per tile.
// 5x5 reg kernels done on VALU from the same LDS patch (cheap).
// All reductions are deterministic (fixed partial slots + ordered finalize).

#define IMG_W 768
#define IMG_H 768
#define HW (IMG_W * IMG_H)
#define NCH 3
#define PST 34           // LDS patch row stride: even (8B-aligned pairs), bank-safe
#define KPW 48           // padded kernel row width: [16 zeros | 15 taps | 17 zeros]
#define NB_RED 64        // partial-sum blocks per channel
#define N_ITER 100
#define EPSF 1e-12f

typedef __attribute__((ext_vector_type(2))) float v2f;
typedef __attribute__((ext_vector_type(8))) float v8f;

// 15x15 'same' correlation of one 16x16 tile via WMMA f32 16x16x4.
// sP: 30-row patch, row 0 = y0-7, col 0 = x0-7, stride PST.
// sKp: padded kernel, sKp[dy*KPW + 16 + dx] = W(dy,dx), zero for dx outside [0,15).
// A layout (16x4 f32): lane L -> M = L%16, K base = 2*(L/16).
// B layout (4x16 f32): lane L -> N = L%16, K base = 2*(L/16).
// D layout: lane L, elem v -> M = v + 8*(L/16), N = L%16.
// B[k,n] = W(dy, 4*ch + k - n); padded buffer makes the fetch unconditional.
__device__ __forceinline__ v8f wmma_conv15(const float* sP, const float* sKp, int lane) {
  const int n  = lane & 15;
  const int kb = (lane >> 4) << 1;
  v8f acc = {0.f, 0.f, 0.f, 0.f, 0.f, 0.f, 0.f, 0.f};
#pragma unroll
  for (int dy = 0; dy < 15; ++dy) {
    const float* prow = &sP[(n + dy) * PST + kb];        // A fragment base
    const float* krow = &sKp[dy * KPW + 16 + kb - n];    // B fragment base (in-bounds)
#pragma unroll
    for (int ch = 0; ch < 8; ++ch) {
      v2f a, b;
      a.x = prow[4 * ch];
      a.y = prow[4 * ch + 1];
      b.x = krow[4 * ch];
      b.y = krow[4 * ch + 1];
      acc = __builtin_amdgcn_wmma_f32_16x16x4_f32(
          /*neg_a=*/false, a, /*neg_b=*/false, b,
          /*c_mod=*/(short)0, acc, /*reuse_a=*/false, /*reuse_b=*/false);
    }
  }
  return acc;
}

// Pass 1: t0 = corr(src,K)  (setup: t0 = src - corr(src,K), also x = src)
//         t1 = corr(src,R0), t2 = corr(src,R1)
__global__ __launch_bounds__(32)
void conv_fwd_kernel(const float* __restrict__ src,
                     const float* __restrict__ kern15,
                     const float* __restrict__ regk,
                     float* __restrict__ t0, float* __restrict__ t1,
                     float* __restrict__ t2, float* __restrict__ xplane,
                     int setup) {
  __shared__ float sP[30 * PST];
  __shared__ float sK[15 * KPW];
  __shared__ float sR[64];
  const int lane = threadIdx.x;
  const int c = blockIdx.z;
  const int y0 = blockIdx.y * 16, x0 = blockIdx.x * 16;
  const float* sp = src + c * HW;

  for (int i = lane; i < 15 * KPW; i += 32) {
    const int dy = i / KPW, dx = i % KPW - 16;
    sK[i] = (dx >= 0 && dx < 15) ? kern15[dy * 15 + dx] : 0.f;
  }
  for (int i = lane; i < 64; i += 32) sR[i] = (i < 50) ? regk[i] : 0.f;
  for (int rr = 0; rr < 30; ++rr) {
    const int gy = y0 + rr - 7, gx = x0 + lane - 7;
    float v = 0.f;
    if (gy >= 0 && gy < IMG_H && gx >= 0 && gx < IMG_W) v = sp[gy * IMG_W + gx];
    sP[rr * PST + lane] = v;
  }
  __syncthreads();

  const v8f acc = wmma_conv15(sP, sK, lane);

  const int n = lane & 15;
  const int mh = (lane >> 4) * 8;
  float a0[8], a1[8];
#pragma unroll
  for (int v = 0; v < 8; ++v) { a0[v] = 0.f; a1[v] = 0.f; }
  for (int dy = 0; dy < 5; ++dy)
    for (int dx = 0; dx < 5; ++dx) {
      const float w0 = sR[dy * 5 + dx];
      const float w1 = sR[25 + dy * 5 + dx];
#pragma unroll
      for (int v = 0; v < 8; ++v) {
        const float pv = sP[(mh + v + dy + 5) * PST + (n + dx + 5)];
        a0[v] += w0 * pv;
        a1[v] += w1 * pv;
      }
    }

#pragma unroll
  for (int v = 0; v < 8; ++v) {
    const int m = mh + v;
    const int idx = c * HW + (y0 + m) * IMG_W + (x0 + n);
    const float kv = acc[v];
    if (setup) {
      const float iv = sP[(m + 7) * PST + (n + 7)];  // src pixel from LDS
      t0[idx] = iv - kv;
      xplane[idx] = iv;
    } else {
      t0[idx] = kv;
    }
    t1[idx] = a0[v];
    t2[idx] = a1[v];
  }
}

// Pass 2 (adjoint): out = corrT(t0,K) + sgn*(w0^2*corrT(t1,R0) + w1^2*corrT(t2,R1))
// corrT == correlation with flipped kernel (flip done while staging to LDS).
__global__ __launch_bounds__(32)
void conv_adj_kernel(const float* __restrict__ t0, const float* __restrict__ t1,
                     const float* __restrict__ t2,
                     const float* __restrict__ kern15,
                     const float* __restrict__ regk,
                     const float* __restrict__ regw,
                     float* __restrict__ out0, float* __restrict__ out1,
                     float sgn) {
  __shared__ float sP0[30 * PST];
  __shared__ float sP1[20 * PST];
  __shared__ float sP2[20 * PST];
  __shared__ float sK[15 * KPW];
  __shared__ float sR[64];
  const int lane = threadIdx.x;
  const int c = blockIdx.z;
  const int y0 = blockIdx.y * 16, x0 = blockIdx.x * 16;

  for (int i = lane; i < 15 * KPW; i += 32) {
    const int dy = i / KPW, dx = i % KPW - 16;
    sK[i] = (dx >= 0 && dx < 15) ? kern15[(14 - dy) * 15 + (14 - dx)] : 0.f;
  }
  for (int i = lane; i < 64; i += 32) {
    float v = 0.f;
    if (i < 50) {
      const int j = i / 25, t = i % 25;
      const int dy = t / 5, dx = t % 5;
      v = regk[j * 25 + (4 - dy) * 5 + (4 - dx)];
    }
    sR[i] = v;
  }
  {
    const float* s = t0 + c * HW;
    for (int rr = 0; rr < 30; ++rr) {
      const int gy = y0 + rr - 7, gx = x0 + lane - 7;
      float v = 0.f;
      if (gy >= 0 && gy < IMG_H && gx >= 0 && gx < IMG_W) v = s[gy * IMG_W + gx];
      sP0[rr * PST + lane] = v;
    }
  }
  {
    const float* s1 = t1 + c * HW;
    const float* s2 = t2 + c * HW;
    for (int rr = 0; rr < 20; ++rr) {
      const int gy = y0 + rr - 2, gx = x0 + lane - 2;
      float v1 = 0.f, v2 = 0.f;
      if (gy >= 0 && gy < IMG_H && gx >= 0 && gx < IMG_W) {
        v1 = s1[gy * IMG_W + gx];
        v2 = s2[gy * IMG_W + gx];
      }
      sP1[rr * PST + lane] = v1;
      sP2[rr * PST + lane] = v2;
    }
  }
  __syncthreads();

  const v8f acc = wmma_conv15(sP0, sK, lane);

  const int n = lane & 15;
  const int mh = (lane >> 4) * 8;
  float a0[8], a1[8];
#pragma unroll
  for (int v = 0; v < 8; ++v) { a0[v] = 0.f; a1[v] = 0.f; }
  for (int dy = 0; dy < 5; ++dy)
    for (int dx = 0; dx < 5; ++dx) {
      const float w0 = sR[dy * 5 + dx];
      const float w1 = sR[25 + dy * 5 + dx];
#pragma unroll
      for (int v = 0; v < 8; ++v) {
        a0[v] += w0 * sP1[(mh + v + dy) * PST + (n + dx)];
        a1[v] += w1 * sP2[(mh + v + dy) * PST + (n + dx)];
      }
    }

  const float s0 = sgn * regw[0] * regw[0];
  const float s1 = sgn * regw[1] * regw[1];
#pragma unroll
  for (int v = 0; v < 8; ++v) {
    const int idx = c * HW + (y0 + mh + v) * IMG_W + (x0 + n);
    const float o = acc[v] + s0 * a0[v] + s1 * a1[v];
    out0[idx] = o;
    if (out1) out1[idx] = o;
  }
}

// Deterministic per-channel partial dot products: partials[c*NB_RED + blk]
__global__ __launch_bounds__(256)
void reduce_dot_kernel(const float* __restrict__ a, const float* __restrict__ b,
                       float* __restrict__ partials) {
  __shared__ float sm[256];
  const int c = blockIdx.y, t = threadIdx.x;
  const float* pa = a + c * HW;
  const float* pb = b + c * HW;
  float s = 0.f;
  for (int i = blockIdx.x * blockDim.x + t; i < HW; i += gridDim.x * blockDim.x)
    s += pa[i] * pb[i];
  sm[t] = s;
  __syncthreads();
  for (int o = 128; o > 0; o >>= 1) {
    if (t < o) sm[t] += sm[t + o];
    __syncthreads();
  }
  if (t == 0) partials[c * NB_RED + blockIdx.x] = sm[0];
}

// scal layout: [0..2]=rtr  [3..5]=alpha  [6..8]=beta
// mode 0: rtr=sum   mode 1: alpha=rtr/(sum+eps)   mode 2: beta=sum/(rtr+eps); rtr=sum
__global__ __launch_bounds__(32)
void finalize_kernel(const float* __restrict__ partials, float* __restrict__ scal,
                     int mode) {
  const int c = threadIdx.x;
  if (c < NCH) {
    float s = 0.f;
    for (int i = 0; i < NB_RED; ++i) s += partials[c * NB_RED + i];
    if (mode == 0) {
      scal[c] = s;
    } else if (mode == 1) {
      scal[3 + c] = scal[c] / (s + EPSF);
    } else {
      scal[6 + c] = s / (scal[c] + EPSF);
      scal[c] = s;
    }
  }
}

// x += alpha*p ; r -= alpha*Ap ; partials of ||r_new||^2 (becomes next rtr)
__global__ __launch_bounds__(256)
void update_xr_kernel(float* __restrict__ x, float* __restrict__ r,
                      const float* __restrict__ p, const float* __restrict__ Ap,
                      const float* __restrict__ scal, float* __restrict__ partials) {
  __shared__ float sm[256];
  const int c = blockIdx.y, t = threadIdx.x;
  const float alpha = scal[3 + c];
  const int base = c * HW;
  float s = 0.f;
  for (int i = blockIdx.x * blockDim.x + t; i < HW; i += gridDim.x * blockDim.x) {
    const int j = base + i;
    x[j] = x[j] + alpha * p[j];
    const float rn = r[j] - alpha * Ap[j];
    r[j] = rn;
    s += rn * rn;
  }
  sm[t] = s;
  __syncthreads();
  for (int o = 128; o > 0; o >>= 1) {
    if (t < o) sm[t] += sm[t + o];
    __syncthreads();
  }
  if (t == 0) partials[c * NB_RED + blockIdx.x] = sm[0];
}

__global__ __launch_bounds__(256)
void update_p_kernel(float* __restrict__ p, const float* __restrict__ r,
                     const float* __restrict__ scal) {
  const int c = blockIdx.y;
  const float beta = scal[6 + c];
  const int base = c * HW;
  for (int i = blockIdx.x * blockDim.x + threadIdx.x; i < HW;
       i += gridDim.x * blockDim.x) {
    const int j = base + i;
    p[j] = r[j] + beta * p[j];
  }
}

extern "C" void kernel_launch(void* const* d_in, const int* in_sizes, int n_in,
                              void* d_out, int out_size, void* d_ws, size_t ws_size,
                              hipStream_t stream) {
  (void)in_sizes; (void)n_in; (void)out_size; (void)ws_size;
  const float* image = (const float*)d_in[0];   // [3,768,768]
  const float* kern  = (const float*)d_in[1];   // [15,15] normalized
  const float* regk  = (const float*)d_in[2];   // [2,5,5]
  const float* regw  = (const float*)d_in[3];   // [2]

  float* out = (float*)d_out;                   // [3(xrp),3,768,768]
  float* x = out;
  float* r = out + (size_t)NCH * HW;
  float* p = out + (size_t)2 * NCH * HW;

  float* ws = (float*)d_ws;
  float* t0 = ws;                               // corr(.,K) field / setup residual
  float* t1 = ws + (size_t)NCH * HW;            // corr(.,R0)
  float* t2 = ws + (size_t)2 * NCH * HW;        // corr(.,R1)
  float* Ap = ws + (size_t)3 * NCH * HW;
  float* partials = ws + (size_t)4 * NCH * HW;  // [3][NB_RED]
  float* scal = partials + NCH * NB_RED;        // rtr[3], alpha[3], beta[3]

  const dim3 cgrid(IMG_W / 16, IMG_H / 16, NCH), cblk(32);
  const dim3 rgrid(NB_RED, NCH), rblk(256);

  // ---- setup: x = image ; r = p = KTb - A(image) ; rtr = ||r||^2 ----
  conv_fwd_kernel<<<cgrid, cblk, 0, stream>>>(image, kern, regk, t0, t1, t2, x, 1);
  conv_adj_kernel<<<cgrid, cblk, 0, stream>>>(t0, t1, t2, kern, regk, regw, r, p, -1.f);
  reduce_dot_kernel<<<rgrid, rblk, 0, stream>>>(r, r, partials);
  finalize_kernel<<<1, 32, 0, stream>>>(partials, scal, 0);

  // ---- 100 CG iterations ----
  for (int it = 0; it < N_ITER; ++it) {
    conv_fwd_kernel<<<cgrid, cblk, 0, stream>>>(p, kern, regk, t0, t1, t2, nullptr, 0);
    conv_adj_kernel<<<cgrid, cblk, 0, stream>>>(t0, t1, t2, kern, regk, regw, Ap, nullptr, 1.f);
    reduce_dot_kernel<<<rgrid, rblk, 0, stream>>>(p, Ap, partials);
    finalize_kernel<<<1, 32, 0, stream>>>(partials, scal, 1);        // alpha
    update_xr_kernel<<<rgrid, rblk, 0, stream>>>(x, r, p, Ap, scal, partials);
    finalize_kernel<<<1, 32, 0, stream>>>(partials, scal, 2);        // beta, rtr
    update_p_kernel<<<rgrid, rblk, 0, stream>>>(p, r, scal);
  }
}